// ShortTermSTGNNEncoder_90606630076475
// MI455X (gfx1250) — compile-verified
//
#include <hip/hip_runtime.h>

// ---- problem constants (from reference) ----
#define B_   4
#define T_   12
#define N_   10000
#define FIN_ 16
#define H_   64
#define E_   160000
#define L_   2
#define BT_  (B_ * T_)         // 48
#define ROWS_ (BT_ * N_)       // 480000 rows of h
#define GR_  (B_ * N_)         // 40000 GRU sequences

typedef float v2f __attribute__((ext_vector_type(2)));
typedef float v8f __attribute__((ext_vector_type(8)));

// D = A(16x4 f32) * B(4x16 f32) + C  -- exact fp32 on the matrix pipe
__device__ __forceinline__ v8f wmma4(v2f a, v2f b, v8f c) {
    return __builtin_amdgcn_wmma_f32_16x16x4_f32(
        /*neg_a=*/false, a, /*neg_b=*/false, b,
        /*c_mod=*/(short)0, c, /*reuse_a=*/false, /*reuse_b=*/false);
}

// ---------------- utility ----------------
__global__ void fill_zero_kernel(float* __restrict__ p, int n) {
    int i = blockIdx.x * 256 + threadIdx.x;
    if (i < n) p[i] = 0.0f;
}

// deg[dst] += w  (GCN symmetric norm, step 1)
__global__ void deg_kernel(const int* __restrict__ ei, const float* __restrict__ ew,
                           float* __restrict__ deg) {
    int e = blockIdx.x * 256 + threadIdx.x;
    if (e < E_) atomicAdd(&deg[ei[E_ + e]], ew[e]);
}

// w_norm[e] = dinv[src] * w * dinv[dst]
__global__ void norm_kernel(const int* __restrict__ ei, const float* __restrict__ ew,
                            const float* __restrict__ deg, float* __restrict__ wn) {
    int e = blockIdx.x * 256 + threadIdx.x;
    if (e >= E_) return;
    int s = ei[e], d = ei[E_ + e];
    float ds = deg[s], dd = deg[d];
    float is = (ds > 0.0f) ? rsqrtf(ds) : 0.0f;
    float id = (dd > 0.0f) ? rsqrtf(dd) : 0.0f;
    wn[e] = is * ew[e] * id;
}

// ---------------- preprocess: h = relu(x @ W_pre + b_pre) ----------------
// one wave = 16 rows x 64 cols, K=16 (4 WMMA k-steps x 4 col tiles)
// W_pre staged once per block into LDS, transposed to [col][k] so each
// B-fragment is a single 8-byte ds_load_b64.
__global__ void pre_gemm_kernel(const float* __restrict__ x, const float* __restrict__ Wp,
                                const float* __restrict__ bp, float* __restrict__ h) {
    __shared__ float wls[H_ * FIN_];   // [col][k], 4 KB
    for (int i = threadIdx.x; i < FIN_ * H_; i += 256) {
        const int k = i >> 6, col = i & 63;          // coalesced read of Wp[k][col]
        wls[col * FIN_ + k] = Wp[i];
    }
    __syncthreads();

    const int wave = (blockIdx.x * blockDim.x + threadIdx.x) >> 5;
    const int lane = threadIdx.x & 31;
    const int l15 = lane & 15, kh = lane >> 4;
    const int rowbase = wave * 16;
    const float* arow = x + (size_t)(rowbase + l15) * FIN_;

    v8f acc[4] = {};
#pragma unroll
    for (int k0 = 0; k0 < FIN_; k0 += 4) {
        const int k = k0 + 2 * kh;
        v2f a = *(const v2f*)(arow + k);
#pragma unroll
        for (int nt = 0; nt < 4; ++nt) {
            const int col = nt * 16 + l15;
            v2f b = *(const v2f*)(wls + col * FIN_ + k);
            acc[nt] = wmma4(a, b, acc[nt]);
        }
    }
#pragma unroll
    for (int nt = 0; nt < 4; ++nt) {
        const int col = nt * 16 + l15;
        const float bias = bp[col];
#pragma unroll
        for (int v = 0; v < 8; ++v) {
            const int row = rowbase + v + 8 * kh;
            float val = acc[nt][v] + bias;
            h[(size_t)row * H_ + col] = (val > 0.0f) ? val : 0.0f;
        }
    }
}

// ---------------- GraphConv dual GEMM ----------------
// reads h once; m = h @ W_lin ; agg = h @ W_root + b_conv (agg is scatter target)
// Both weight matrices staged transposed into LDS (32 KB/block).
__global__ void conv_gemm_kernel(const float* __restrict__ h, const float* __restrict__ Wl,
                                 const float* __restrict__ Wr, const float* __restrict__ bc,
                                 float* __restrict__ m, float* __restrict__ agg) {
    __shared__ float wl_s[H_ * H_];    // [col][k], 16 KB
    __shared__ float wr_s[H_ * H_];    // [col][k], 16 KB
    for (int i = threadIdx.x; i < H_ * H_; i += 256) {
        const int k = i >> 6, col = i & 63;
        wl_s[col * H_ + k] = Wl[i];
        wr_s[col * H_ + k] = Wr[i];
    }
    __syncthreads();

    const int wave = (blockIdx.x * blockDim.x + threadIdx.x) >> 5;
    const int lane = threadIdx.x & 31;
    const int l15 = lane & 15, kh = lane >> 4;
    const int rowbase = wave * 16;
    const float* arow = h + (size_t)(rowbase + l15) * H_;

    v8f accM[4] = {};
    v8f accR[4] = {};
#pragma unroll
    for (int k0 = 0; k0 < H_; k0 += 4) {
        const int k = k0 + 2 * kh;
        v2f a = *(const v2f*)(arow + k);
#pragma unroll
        for (int nt = 0; nt < 4; ++nt) {
            const int col = nt * 16 + l15;
            v2f b1 = *(const v2f*)(wl_s + col * H_ + k);
            accM[nt] = wmma4(a, b1, accM[nt]);
            v2f b2 = *(const v2f*)(wr_s + col * H_ + k);
            accR[nt] = wmma4(a, b2, accR[nt]);
        }
    }
#pragma unroll
    for (int nt = 0; nt < 4; ++nt) {
        const int col = nt * 16 + l15;
        const float bias = bc[col];
#pragma unroll
        for (int v = 0; v < 8; ++v) {
            const size_t row = (size_t)(rowbase + v + 8 * kh);
            m[row * H_ + col]   = accM[nt][v];
            agg[row * H_ + col] = accR[nt][v] + bias;
        }
    }
}

// ---------------- edge gather/scale/scatter-add ----------------
// 16 threads per edge (float4 each); edge metadata loaded once, loop over BT slices.
// agg (123 MB) and m (123 MB) are L2-resident on MI455X (192 MB L2).
__global__ void edge_scatter_kernel(const float* __restrict__ m, const float* __restrict__ wn,
                                    const int* __restrict__ ei, float* __restrict__ agg) {
    const int t = blockIdx.x * 256 + threadIdx.x;
    const int e = t >> 4;
    const int c = (t & 15) * 4;
    const int s = ei[e], d = ei[E_ + e];
    const float w = wn[e];
    const float* ms = m + (size_t)s * H_ + c;
    float* ad = agg + (size_t)d * H_ + c;
    const size_t slice = (size_t)N_ * H_;
    for (int bt = 0; bt < BT_; ++bt) {
        const float4 v = *(const float4*)(ms + bt * slice);
        float* a = ad + bt * slice;
        atomicAdd(a + 0, v.x * w);
        atomicAdd(a + 1, v.y * w);
        atomicAdd(a + 2, v.z * w);
        atomicAdd(a + 3, v.w * w);
    }
}

// h = relu(agg), vectorized
__global__ void relu_copy_kernel(const float4* __restrict__ agg, float4* __restrict__ h) {
    int i = blockIdx.x * 256 + threadIdx.x;
    float4 v = agg[i];
    v.x = fmaxf(v.x, 0.0f);
    v.y = fmaxf(v.y, 0.0f);
    v.z = fmaxf(v.z, 0.0f);
    v.w = fmaxf(v.w, 0.0f);
    h[i] = v;
}

// ---------------- GRU step t ----------------
// One column tile per block; 4 waves each own a 16x16 tile of h_new at
// consecutive row tiles. The block stages the 16-column slice of W_ih/W_hh
// (3 gates x 16 cols x 64 k = 12 KB each) into LDS; W_ih is stored [col][k]
// which is exactly the B-fragment layout -> ds_load_b64 per fragment.
// 96 WMMAs per wave, then gate math in VALU. seq transpose by addressing.
__global__ void __launch_bounds__(128)
gru_step_kernel(const float* __restrict__ seq, const float* __restrict__ hprev,
                const float* __restrict__ Wih, const float* __restrict__ Whh,
                const float* __restrict__ bih, const float* __restrict__ bhh,
                int t, float* __restrict__ hout) {
    __shared__ float wih_s[3 * 16 * H_];   // [(g*16+cl)][k], 12 KB
    __shared__ float whh_s[3 * 16 * H_];   // 12 KB

    const int colTile = blockIdx.x & 3;
    const int colb = colTile * 16;
    for (int i = threadIdx.x; i < 3 * 16 * H_; i += 128) {
        const int gcl = i >> 6, k = i & 63;
        const int g = gcl >> 4, cl = gcl & 15;
        const int col = g * H_ + colb + cl;
        wih_s[i] = Wih[(size_t)col * H_ + k];   // coalesced: contiguous k per row
        whh_s[i] = Whh[(size_t)col * H_ + k];
    }
    __syncthreads();

    const int lane = threadIdx.x & 31;
    const int l15 = lane & 15, kh = lane >> 4;
    const int rowTile = (blockIdx.x >> 2) * 4 + (threadIdx.x >> 5);
    const int rowbase = rowTile * 16;

    const int row_a = rowbase + l15;          // GRU row = b*N + n
    const int bb = row_a / N_;
    const int nn = row_a - bb * N_;
    const float* xrow = seq + ((size_t)(bb * T_ + t) * N_ + nn) * H_;  // x_t for this row
    const float* hrow = hprev + (size_t)row_a * H_;

    v8f gi[3] = {};
    v8f gh[3] = {};
#pragma unroll
    for (int k0 = 0; k0 < H_; k0 += 4) {
        const int k = k0 + 2 * kh;
        v2f ax = *(const v2f*)(xrow + k);
        v2f ah = *(const v2f*)(hrow + k);
#pragma unroll
        for (int g = 0; g < 3; ++g) {
            v2f bi = *(const v2f*)(wih_s + (g * 16 + l15) * H_ + k);
            gi[g] = wmma4(ax, bi, gi[g]);
            v2f bh = *(const v2f*)(whh_s + (g * 16 + l15) * H_ + k);
            gh[g] = wmma4(ah, bh, gh[g]);
        }
    }

    const int col = colb + l15;
    const float bir = bih[col], biz = bih[H_ + col], bin = bih[2 * H_ + col];
    const float bhr = bhh[col], bhz = bhh[H_ + col], bhn = bhh[2 * H_ + col];
#pragma unroll
    for (int v = 0; v < 8; ++v) {
        const int row = rowbase + v + 8 * kh;
        const float ir = gi[0][v] + bir, iz = gi[1][v] + biz, in_ = gi[2][v] + bin;
        const float hr = gh[0][v] + bhr, hz = gh[1][v] + bhz, hn = gh[2][v] + bhn;
        const float r  = 1.0f / (1.0f + __expf(-(ir + hr)));
        const float z  = 1.0f / (1.0f + __expf(-(iz + hz)));
        const float ng = tanhf(in_ + r * hn);
        const float hp = hprev[(size_t)row * H_ + col];
        hout[(size_t)row * H_ + col] = (1.0f - z) * ng + z * hp;
    }
}

// ---------------- driver ----------------
extern "C" void kernel_launch(void* const* d_in, const int* in_sizes, int n_in,
                              void* d_out, int out_size, void* d_ws, size_t ws_size,
                              hipStream_t stream) {
    (void)in_sizes; (void)n_in; (void)out_size; (void)ws_size;
    const float* x   = (const float*)d_in[0];
    const int*   ei  = (const int*)d_in[1];   // [2, E]
    const float* ew  = (const float*)d_in[2];
    const float* Wp  = (const float*)d_in[3];
    const float* bp  = (const float*)d_in[4];
    const float* Wl  = (const float*)d_in[5]; // [L, H, H]
    const float* Wr  = (const float*)d_in[6];
    const float* bc  = (const float*)d_in[7]; // [L, H]
    const float* Wih = (const float*)d_in[8]; // [3H, H]
    const float* Whh = (const float*)d_in[9];
    const float* bih = (const float*)d_in[10];
    const float* bhh = (const float*)d_in[11];
    float* out = (float*)d_out;

    // workspace carve-up (~390 MB)
    float* h    = (float*)d_ws;                    // ROWS_*H_
    float* m    = h   + (size_t)ROWS_ * H_;        // ROWS_*H_
    float* agg  = m   + (size_t)ROWS_ * H_;        // ROWS_*H_
    float* wn   = agg + (size_t)ROWS_ * H_;        // E_
    float* deg  = wn  + E_;                        // N_
    float* gA   = deg + N_;                        // GR_*H_
    float* gB   = gA  + (size_t)GR_ * H_;          // GR_*H_

    // edge normalization
    fill_zero_kernel<<<(N_ + 255) / 256, 256, 0, stream>>>(deg, N_);
    fill_zero_kernel<<<(GR_ * H_ + 255) / 256, 256, 0, stream>>>(gA, GR_ * H_);
    deg_kernel<<<E_ / 256, 256, 0, stream>>>(ei, ew, deg);
    norm_kernel<<<E_ / 256, 256, 0, stream>>>(ei, ew, deg, wn);

    // preprocess: 30000 row tiles, 8 waves/block
    pre_gemm_kernel<<<ROWS_ / 16 / 8, 256, 0, stream>>>(x, Wp, bp, h);

    // GraphConv layers
    for (int l = 0; l < L_; ++l) {
        conv_gemm_kernel<<<ROWS_ / 16 / 8, 256, 0, stream>>>(
            h, Wl + l * H_ * H_, Wr + l * H_ * H_, bc + l * H_, m, agg);
        edge_scatter_kernel<<<E_ * 16 / 256, 256, 0, stream>>>(m, wn, ei, agg);
        relu_copy_kernel<<<ROWS_ * H_ / 4 / 256, 256, 0, stream>>>(
            (const float4*)agg, (float4*)h);
    }

    // GRU over T steps; grid = 625 row groups x 4 col tiles, 4 waves/block
    const float* prev = gA;
    for (int t = 0; t < T_; ++t) {
        float* dst = (t == T_ - 1) ? out : ((t & 1) ? gA : gB);
        gru_step_kernel<<<(GR_ / 16 / 4) * 4, 128, 0, stream>>>(
            h, prev, Wih, Whh, bih, bhh, t, dst);
        prev = dst;
    }
}